// ThreeLayerSparseConv_20564303414103
// MI455X (gfx1250) — compile-verified
//
#include <hip/hip_runtime.h>

#define N_VOX 100000
#define C_CH  64
#define K_OFF 27
#define E_ENT 50000
#define BN_EPS 1e-5f

// LDS strides (floats / dwords), chosen for conflict-free fragment reads
#define AS  68   // A tile row stride: banks (4m+kh)%64 distinct across the wave, 16B-aligned rows
#define BS2 160  // B tile k-pair group stride: half-wave bank ranges 0..31 vs 32..63 disjoint

#define TILE_E 64

typedef float v2f __attribute__((ext_vector_type(2)));
typedef float v8f __attribute__((ext_vector_type(8)));

__global__ void zero_kernel(float* __restrict__ p, int n) {
    int i = blockIdx.x * blockDim.x + threadIdx.x;
    int stride = gridDim.x * blockDim.x;
    for (; i < n; i += stride) p[i] = 0.0f;
}

// One block: kernel-offset k (blockIdx.y), 64 rulebook entries (blockIdx.x*64), all 64 out channels.
// 8 wave32s: wave>>1 -> E-subtile (16 rows), wave&1 -> 32-channel subtile.
// W is staged in LDS in fragment-native order: B2[kp][n][j] = W[2kp+j][n], so every
// B fragment is a single contiguous ds_load_b64 (no register marshalling movs).
// Per k-step: 3 ds_load_b64 feed 2 v_wmma_f32_16x16x4_f32 (A fragment reused).
__global__ __launch_bounds__(256) void conv_kernel(
    const float* __restrict__ x,       // [N_VOX, 64]
    const float* __restrict__ W,       // [K_OFF, 64, 64]
    const int*   __restrict__ in_idx,  // [K_OFF, E_ENT]
    const int*   __restrict__ out_idx, // [K_OFF, E_ENT]
    float* __restrict__ out)           // [N_VOX, 64], pre-zeroed; scatter-add
{
    __shared__ __align__(16) float A[TILE_E * AS];
    __shared__ __align__(16) float B2[32 * BS2];  // 32 k-pair groups

    const int k  = blockIdx.y;
    const int e0 = blockIdx.x * TILE_E;
    const int t  = threadIdx.x;

    // Stage W_k (64x64) into LDS in k-pair-major fragment order, coalesced global reads
    const float* Wk = W + (size_t)k * 4096;
    #pragma unroll
    for (int i = 0; i < 16; ++i) {
        int idx  = t + i * 256;
        int krow = idx >> 6;
        int n    = idx & 63;
        B2[(krow >> 1) * BS2 + n * 2 + (krow & 1)] = Wk[idx];
    }

    // Gather 64 rows of x into LDS: 4 threads per row, 16 floats each (4x float4)
    {
        int r  = t >> 2;
        int cg = t & 3;  // 16-float column group
        int e  = e0 + r;
        float4 v0 = make_float4(0.f, 0.f, 0.f, 0.f);
        float4 v1 = v0, v2 = v0, v3 = v0;
        if (e < E_ENT) {
            int row = in_idx[(size_t)k * E_ENT + e];
            const float4* src = (const float4*)(x + (size_t)row * 64 + cg * 16);
            v0 = src[0]; v1 = src[1]; v2 = src[2]; v3 = src[3];
        }
        float4* dst = (float4*)(A + r * AS + cg * 16);
        dst[0] = v0; dst[1] = v1; dst[2] = v2; dst[3] = v3;
    }
    __syncthreads();

    const int wave = t >> 5;
    const int lane = t & 31;
    const int em = wave >> 1;       // which 16 entries (0..3)
    const int nh = (wave & 1) * 32; // which 32 output channels (0 or 32)
    const int ml = lane & 15;       // M index (A) / N index (B, D)
    const int kh = (lane >> 4) * 2; // half-wave K offset {0,2}

    v8f acc0 = {0.f, 0.f, 0.f, 0.f, 0.f, 0.f, 0.f, 0.f};
    v8f acc1 = acc0;
    const float* arow  = A + (em * 16 + ml) * AS + kh;            // A[m][kk+kh .. +1]
    const float* bbase = B2 + (kh >> 1) * BS2 + (nh + ml) * 2;    // {W[k][n], W[k+1][n]} pairs

    #pragma unroll
    for (int kk = 0; kk < 64; kk += 4) {
        // A frag 16x4: lanes 0-15 M=0..15 K={kk,kk+1}; lanes 16-31 K={kk+2,kk+3}
        v2f a = *(const v2f*)(arow + kk);
        // B frags 4x16 as contiguous k-pairs: b64 each; b0/b1 pairable via 2addr
        const float* bp = bbase + (kk >> 1) * BS2;
        v2f b0 = *(const v2f*)(bp);
        v2f b1 = *(const v2f*)(bp + 32);
        acc0 = __builtin_amdgcn_wmma_f32_16x16x4_f32(
            false, a, false, b0, (short)0, acc0, false, false);
        acc1 = __builtin_amdgcn_wmma_f32_16x16x4_f32(
            false, a, false, b1, (short)0, acc1, false, false);
    }

    // D 16x16 f32 layout: VGPR r -> M=r (lanes 0-15) / M=r+8 (lanes 16-31); lane&15 -> N
    const int mh = (lane >> 4) * 8;
    #pragma unroll
    for (int r = 0; r < 8; ++r) {
        int m = mh + r;
        int e = e0 + em * 16 + m;
        if (e < E_ENT) {
            int row = out_idx[(size_t)k * E_ENT + e];
            float* orow = out + (size_t)row * 64 + nh + ml;
            atomicAdd(orow, acc0[r]);
            atomicAdd(orow + 16, acc1[r]);
        }
    }
}

// Per-channel sum / sum-of-squares into stats[0..63] / stats[64..127]
__global__ __launch_bounds__(256) void stats_kernel(
    const float* __restrict__ x, float* __restrict__ stats)
{
    __shared__ float s_sum[256];
    __shared__ float s_sq[256];
    const int t  = threadIdx.x;
    const int c  = t & 63;
    const int rl = t >> 6;  // 4 row-lanes per channel

    float sum = 0.f, sq = 0.f;
    for (int r = blockIdx.x * 4 + rl; r < N_VOX; r += gridDim.x * 4) {
        float v = x[(size_t)r * 64 + c];
        sum += v;
        sq  += v * v;
    }
    s_sum[t] = sum;
    s_sq[t]  = sq;
    __syncthreads();
    if (rl == 0) {
        sum = s_sum[c] + s_sum[64 + c] + s_sum[128 + c] + s_sum[192 + c];
        sq  = s_sq[c]  + s_sq[64 + c]  + s_sq[128 + c]  + s_sq[192 + c];
        atomicAdd(stats + c, sum);
        atomicAdd(stats + 64 + c, sq);
    }
}

// y = relu((x - mean) * rsqrt(var + eps) * gamma + beta), vectorized float4
__global__ __launch_bounds__(256) void bnrelu_kernel(
    const float* __restrict__ x, const float* __restrict__ stats,
    const float* __restrict__ gamma, const float* __restrict__ beta,
    float* __restrict__ out)
{
    const float invN = 1.0f / (float)N_VOX;
    long i4 = (long)blockIdx.x * blockDim.x + threadIdx.x;
    const long stride = (long)gridDim.x * blockDim.x;
    const long total4 = (long)N_VOX * C_CH / 4;
    for (; i4 < total4; i4 += stride) {
        const int c0 = (int)((i4 * 4) & 63);
        float4 v = ((const float4*)x)[i4];
        float vv[4] = {v.x, v.y, v.z, v.w};
        float oo[4];
        #pragma unroll
        for (int j = 0; j < 4; ++j) {
            int cc = c0 + j;
            float m   = stats[cc] * invN;
            float var = stats[64 + cc] * invN - m * m;
            float sc  = gamma[cc] * __frsqrt_rn(var + BN_EPS);
            float y   = (vv[j] - m) * sc + beta[cc];
            oo[j] = y > 0.f ? y : 0.f;
        }
        ((float4*)out)[i4] = make_float4(oo[0], oo[1], oo[2], oo[3]);
    }
}

extern "C" void kernel_launch(void* const* d_in, const int* in_sizes, int n_in,
                              void* d_out, int out_size, void* d_ws, size_t ws_size,
                              hipStream_t stream) {
    const float* feats   = (const float*)d_in[0];
    const int*   in_idx  = (const int*)d_in[1];
    const int*   out_idx = (const int*)d_in[2];
    const float* W[3]     = {(const float*)d_in[3], (const float*)d_in[4], (const float*)d_in[5]};
    const float* gamma[3] = {(const float*)d_in[6], (const float*)d_in[8], (const float*)d_in[10]};
    const float* beta[3]  = {(const float*)d_in[7], (const float*)d_in[9], (const float*)d_in[11]};

    float* buf0  = (float*)d_ws;                     // conv output [N, 64]
    float* stats = buf0 + (size_t)N_VOX * C_CH;      // 128 floats (sum | sumsq)
    float* buf1  = stats + 128;                      // bn output [N, 64]
    float* out   = (float*)d_out;

    const dim3 cgrid((E_ENT + TILE_E - 1) / TILE_E, K_OFF);
    const long total4 = (long)N_VOX * C_CH / 4;
    const int bn_blocks = (int)((total4 + 255) / 256) < 2048 ? (int)((total4 + 255) / 256) : 2048;

    const float* xin = feats;
    for (int L = 0; L < 3; ++L) {
        // zero conv accumulator + stats slab (contiguous)
        zero_kernel<<<2048, 256, 0, stream>>>(buf0, N_VOX * C_CH + 128);
        conv_kernel<<<cgrid, 256, 0, stream>>>(xin, W[L], in_idx, out_idx, buf0);
        stats_kernel<<<512, 256, 0, stream>>>(buf0, stats);
        float* dst = (L == 2) ? out : buf1;
        bnrelu_kernel<<<bn_blocks, 256, 0, stream>>>(buf0, stats, gamma[L], beta[L], dst);
        xin = buf1;
    }
}